// MSFeatureExtraction_46127948759309
// MI455X (gfx1250) — compile-verified
//
#include <hip/hip_runtime.h>
#include <math.h>

// ---------------- constants ----------------
#define HH 224
#define WW 224
#define L_TOK 50176          // 224*224
#define DM 32                // d_model
#define DI 64                // d_inner
#define DS 16                // d_state
#define DTR 2                // dt_rank
#define NX 34                // xproj out cols (2 + 16 + 16)
#define LDX 48               // padded xdbl leading dim
#define CH 112               // scan chunk length
#define NC 448               // number of chunks (448*112 = 50176)
#define OUTD 2048
#define NHB (OUTD / 16)      // 128 n-tiles in head
#define KSPL 7               // head k-split: 50176/7 = 7168 per block
#define KB 256               // staged k-values per LDS chunk (16 rows x 256 f32 = 16 KB)
#define KCH ((L_TOK / KSPL) / KB)   // 28 chunks per block

typedef float v2f __attribute__((ext_vector_type(2)));
typedef float v8f __attribute__((ext_vector_type(8)));

__device__ __forceinline__ float siluf(float x) { return x / (1.0f + __expf(-x)); }
__device__ __forceinline__ float softplusf(float x) { return (x > 20.0f) ? x : __logf(1.0f + __expf(x)); }
__device__ __forceinline__ void atomAddF(float* p, float v) {
    __hip_atomic_fetch_add(p, v, __ATOMIC_RELAXED, __HIP_MEMORY_SCOPE_AGENT);
}
__device__ __forceinline__ void waitTensor() {
#if __has_builtin(__builtin_amdgcn_s_wait_tensorcnt)
    __builtin_amdgcn_s_wait_tensorcnt(0);
#else
    asm volatile("s_wait_tensorcnt 0x0" ::: "memory");
#endif
}

// ---------------- Tensor Data Mover staging (guarded) ----------------
#if __has_builtin(__builtin_amdgcn_tensor_load_to_lds)
#define HAVE_TDM 1
typedef unsigned int u32x4 __attribute__((ext_vector_type(4)));
typedef int i32x4 __attribute__((ext_vector_type(4)));
typedef int i32x8 __attribute__((ext_vector_type(8)));

// DMA a (rows x cols) f32 tile (row stride rowStrideElems) from global into LDS.
// D# packing per CDNA5 ISA 8.3/8.4 (2D tensor, groups 2/3 unused).
__device__ __forceinline__ void tdm_load_tile(const float* gsrc, void* lds_dst,
                                              int rows, int cols, int rowStrideElems) {
    unsigned long long ga = (unsigned long long)(uintptr_t)gsrc;
    u32x4 g0;
    g0.x = 1u;                                          // count=1 (valid user descriptor)
    g0.y = (unsigned)(uintptr_t)lds_dst;                // lds_addr (bytes)
    g0.z = (unsigned)ga;                                // global_addr[31:0]
    g0.w = ((unsigned)(ga >> 32) & 0x01FFFFFFu) | (2u << 30);  // global_addr[56:32] | type=2
    i32x8 g1;
    g1[0] = (int)(2u << 16);                                        // data_size = 4B
    g1[1] = (int)(((unsigned)rowStrideElems & 0xFFFFu) << 16);      // tensor_dim0[15:0]
    g1[2] = (int)((((unsigned)rowStrideElems >> 16) & 0xFFFFu) |    // tensor_dim0[31:16]
                  (0x0800u << 16));                                 // tensor_dim1[15:0] = 2048
    g1[3] = (int)(((unsigned)cols & 0xFFFFu) << 16);                // tile_dim0
    g1[4] = (int)((unsigned)rows & 0xFFFFu);                        // tile_dim1 (tile_dim2=0)
    g1[5] = (int)rowStrideElems;                                    // tensor_dim0_stride[31:0]
    g1[6] = 0;
    g1[7] = 0;
    i32x4 z4 = {0, 0, 0, 0};
#if __clang_major__ >= 23
    i32x8 z8 = {0, 0, 0, 0, 0, 0, 0, 0};
    __builtin_amdgcn_tensor_load_to_lds(g0, g1, z4, z4, z8, 0);
#else
    __builtin_amdgcn_tensor_load_to_lds(g0, g1, z4, z4, 0);
#endif
}
#else
#define HAVE_TDM 0
#endif

// ---------------- generic fp32 WMMA GEMM ----------------
// C[M,N] = A[M,K] * W^T (+bias).  W is N x K row-major.
// AT: A[m][k] = A[k*lda + m] (channel-major feature maps / token transpose).
// One wave per 16x16 tile, 8 waves per block.  M must be a multiple of 16.
template <int AT>
__global__ void gemm_wmma_k(const float* __restrict__ A, int lda,
                            const float* __restrict__ W, const float* __restrict__ bias,
                            float* __restrict__ C, int ldc, int M, int N, int K) {
    int wave = blockIdx.x * (blockDim.x >> 5) + (threadIdx.x >> 5);
    int MT = M >> 4;
    int NT = (N + 15) >> 4;
    if (wave >= MT * NT) return;                 // wave-uniform exit: EXEC stays all-ones
    int mt = wave / NT, nt = wave % NT;
    int lane = threadIdx.x & 31;
    int half = lane >> 4;
    int lm = lane & 15;
    int m = (mt << 4) + lm;
    int n = (nt << 4) + lm;
    int nw = (n < N) ? n : 0;                    // clamp for padded N (xproj N=34)

    v8f acc = {0.f, 0.f, 0.f, 0.f, 0.f, 0.f, 0.f, 0.f};
    for (int k = 0; k < K; k += 4) {
        int ka = k + (half << 1);                // lane half selects K pair {0,1} vs {2,3}
        v2f a, b;
        if (AT) {
            a.x = A[(size_t)ka * lda + m];
            a.y = A[(size_t)(ka + 1) * lda + m];
        } else {
            a.x = A[(size_t)m * lda + ka];
            a.y = A[(size_t)m * lda + ka + 1];
        }
        b.x = W[(size_t)nw * K + ka];
        b.y = W[(size_t)nw * K + ka + 1];
        acc = __builtin_amdgcn_wmma_f32_16x16x4_f32(false, a, false, b, (short)0, acc, false, false);
    }
    if (n < N) {
        float bv = bias ? bias[n] : 0.0f;
#pragma unroll
        for (int r = 0; r < 8; ++r) {
            int row = (mt << 4) + r + (half << 3);   // D layout: row = r + 8*half
            C[(size_t)row * ldc + n] = acc[r] + bv;
        }
    }
}

// ---------------- head GEMM ----------------
// relu-head input: (32 x 50176) @ out_w^T -> (32 x 2048), both modalities.
// Each block: one 16-column out_w tile + one K-slice. Weight tiles are DMA'd by the
// Tensor Data Mover into a double-buffered LDS tile (s_wait_tensorcnt + barrier);
// all 8 waves consume the same LDS tile for ms/pan x 2 row-tiles, so the 411 MB
// weight is streamed from HBM exactly once. Partial sums land via f32 atomics.
__global__ void __launch_bounds__(256) head_gemm_k(const float* __restrict__ msT,
                                                   const float* __restrict__ panT,
                                                   const float* __restrict__ Wo,
                                                   float* __restrict__ acc) {
    __shared__ float bbuf[2][16][KB];            // 32 KB double buffer
    int nt = blockIdx.x % NHB;
    int ks = blockIdx.x / NHB;
    int n0 = nt * 16;
    int k0 = ks * (L_TOK / KSPL);
    int wv = threadIdx.x >> 5;
    int lane = threadIdx.x & 31;
    int half = lane >> 4;
    int lm = lane & 15;
    int n = n0 + lm;

    v8f cm0 = {0.f, 0.f, 0.f, 0.f, 0.f, 0.f, 0.f, 0.f};
    v8f cm1 = cm0, cp0 = cm0, cp1 = cm0;

    // stage chunk 0
#if HAVE_TDM
    if (wv == 0) {
        tdm_load_tile(Wo + (size_t)n0 * L_TOK + k0, &bbuf[0][0][0], 16, KB, L_TOK);
        waitTensor();
    }
#else
    for (int i = threadIdx.x; i < 16 * KB; i += 256)
        bbuf[0][i / KB][i % KB] = Wo[(size_t)(n0 + i / KB) * L_TOK + k0 + (i % KB)];
#endif
    __syncthreads();

    for (int c = 0; c < KCH; ++c) {
        int cur = c & 1;
        int kbase = k0 + c * KB;
#if HAVE_TDM
        if (wv == 0 && c + 1 < KCH)              // overlap next DMA with compute
            tdm_load_tile(Wo + (size_t)n0 * L_TOK + kbase + KB, &bbuf[cur ^ 1][0][0],
                          16, KB, L_TOK);
#else
        if (c + 1 < KCH)
            for (int i = threadIdx.x; i < 16 * KB; i += 256)
                bbuf[cur ^ 1][i / KB][i % KB] =
                    Wo[(size_t)(n0 + i / KB) * L_TOK + kbase + KB + (i % KB)];
#endif
        int kloc0 = wv * (KB / 8);               // each wave owns 32 staged k-values
#pragma unroll
        for (int kk = 0; kk < KB / 8; kk += 4) {
            int kl = kloc0 + kk + (half << 1);
            int kg = kbase + kl;
            __builtin_prefetch(msT + (size_t)(kg + 64) * DM, 0, 0);
            v2f b, a;
            b.x = bbuf[cur][lm][kl];
            b.y = bbuf[cur][lm][kl + 1];
            a.x = msT[(size_t)kg * DM + lm];        a.y = msT[(size_t)(kg + 1) * DM + lm];
            cm0 = __builtin_amdgcn_wmma_f32_16x16x4_f32(false, a, false, b, (short)0, cm0, false, false);
            a.x = msT[(size_t)kg * DM + 16 + lm];   a.y = msT[(size_t)(kg + 1) * DM + 16 + lm];
            cm1 = __builtin_amdgcn_wmma_f32_16x16x4_f32(false, a, false, b, (short)0, cm1, false, false);
            a.x = panT[(size_t)kg * DM + lm];       a.y = panT[(size_t)(kg + 1) * DM + lm];
            cp0 = __builtin_amdgcn_wmma_f32_16x16x4_f32(false, a, false, b, (short)0, cp0, false, false);
            a.x = panT[(size_t)kg * DM + 16 + lm];  a.y = panT[(size_t)(kg + 1) * DM + 16 + lm];
            cp1 = __builtin_amdgcn_wmma_f32_16x16x4_f32(false, a, false, b, (short)0, cp1, false, false);
        }
#if HAVE_TDM
        if (wv == 0) waitTensor();               // DMA landed before anyone flips buffers
#endif
        __syncthreads();
    }

#pragma unroll
    for (int r = 0; r < 8; ++r) {
        int row = r + (half << 3);
        atomAddF(&acc[(size_t)row * OUTD + n], cm0[r]);
        atomAddF(&acc[(size_t)(16 + row) * OUTD + n], cm1[r]);
        atomAddF(&acc[(size_t)DM * OUTD + (size_t)row * OUTD + n], cp0[r]);
        atomAddF(&acc[(size_t)DM * OUTD + (size_t)(16 + row) * OUTD + n], cp1[r]);
    }
}

__global__ void head_epi_k(const float* __restrict__ acc, const float* __restrict__ bias,
                           float* __restrict__ out) {
    int i = blockIdx.x * blockDim.x + threadIdx.x;
    if (i >= 2 * DM * OUTD) return;
    float v = acc[i] + bias[i % OUTD];
    out[i] = v > 0.0f ? v : 0.0f;
}

__global__ void head_feat_k(const float* __restrict__ out, float* __restrict__ feat) {
    int i = blockIdx.x * blockDim.x + threadIdx.x;
    if (i >= 2 * OUTD) return;
    const float* b = out + (size_t)(i / OUTD) * DM * OUTD;
    int o = i % OUTD;
    float s = 0.0f;
#pragma unroll
    for (int d = 0; d < DM; ++d) s += b[(size_t)d * OUTD + o];
    feat[i] = s * (1.0f / DM);
}

// ---------------- encoder pieces ----------------
__global__ void conv3_k(const float* __restrict__ in, int cin, const float* __restrict__ w,
                        const float* __restrict__ b, float* __restrict__ out, int leaky) {
    int idx = blockIdx.x * blockDim.x + threadIdx.x;
    if (idx >= DM * L_TOK) return;
    int co = idx / L_TOK;
    int p = idx % L_TOK;
    int y = p / WW, x = p % WW;
    float s = b[co];
    for (int ci = 0; ci < cin; ++ci) {
        const float* ip = in + (size_t)ci * L_TOK;
        const float* wp = w + ((size_t)co * cin + ci) * 9;
#pragma unroll
        for (int ky = 0; ky < 3; ++ky) {
            int yy = y + ky - 1;
            if (yy < 0 || yy >= HH) continue;
#pragma unroll
            for (int kx = 0; kx < 3; ++kx) {
                int xx = x + kx - 1;
                if (xx < 0 || xx >= WW) continue;
                s += ip[yy * WW + xx] * wp[ky * 3 + kx];
            }
        }
    }
    if (leaky && s < 0.0f) s *= 0.2f;
    out[idx] = s;
}

// instance norm over H*W for channels 0..15, in place (LDS tree reduce)
__global__ void instnorm_k(float* __restrict__ feat, const float* __restrict__ nw,
                           const float* __restrict__ nb) {
    __shared__ float r1[256], r2[256];
    int c = blockIdx.x;
    float* p = feat + (size_t)c * L_TOK;
    float s = 0.0f, s2 = 0.0f;
    for (int i = threadIdx.x; i < L_TOK; i += blockDim.x) { float v = p[i]; s += v; s2 += v * v; }
    r1[threadIdx.x] = s; r2[threadIdx.x] = s2;
    __syncthreads();
    for (int st = 128; st > 0; st >>= 1) {
        if ((int)threadIdx.x < st) { r1[threadIdx.x] += r1[threadIdx.x + st]; r2[threadIdx.x] += r2[threadIdx.x + st]; }
        __syncthreads();
    }
    float mu = r1[0] * (1.0f / L_TOK);
    float var = r2[0] * (1.0f / L_TOK) - mu * mu;
    float inv = rsqrtf(var + 1e-5f);
    float g = nw[c], bb = nb[c];
    for (int i = threadIdx.x; i < L_TOK; i += blockDim.x) p[i] = (p[i] - mu) * inv * g + bb;
}

// ---------------- token-domain elementwise ----------------
__global__ void layernorm_k(const float* __restrict__ in, const float* __restrict__ w,
                            const float* __restrict__ b, float* __restrict__ out) {
    int t = blockIdx.x * blockDim.x + threadIdx.x;
    if (t >= L_TOK) return;
    const float* p = in + (size_t)t * DM;
    float s = 0.0f;
#pragma unroll
    for (int i = 0; i < DM; ++i) s += p[i];
    float mu = s * (1.0f / DM);
    float v = 0.0f;
#pragma unroll
    for (int i = 0; i < DM; ++i) { float d = p[i] - mu; v += d * d; }
    float inv = rsqrtf(v * (1.0f / DM) + 1e-5f);
    float* o = out + (size_t)t * DM;
#pragma unroll
    for (int i = 0; i < DM; ++i) o[i] = (p[i] - mu) * inv * w[i] + b[i];
}

__global__ void fill_k(float* __restrict__ p, int n, float v) {
    int i = blockIdx.x * blockDim.x + threadIdx.x;
    if (i < n) p[i] = v;
}
__global__ void addip_k(float* __restrict__ dst, const float* __restrict__ src, int n) {
    int i = blockIdx.x * blockDim.x + threadIdx.x;
    if (i < n) dst[i] += src[i];
}
// dst[l][c] = c < DM/2 ? lo[l][c] : hi[l][c]
__global__ void concat_k(float* __restrict__ dst, const float* __restrict__ lo,
                         const float* __restrict__ hi) {
    int i = blockIdx.x * blockDim.x + threadIdx.x;
    if (i >= L_TOK * DM) return;
    dst[i] = ((i % DM) < (DM / 2)) ? lo[i] : hi[i];
}

// ---------------- mamba pieces ----------------
// causal depthwise conv over L (taps 4) + SiLU; xin = cols [0,64) of xz (ld 128)
__global__ void conv1d_silu_k(const float* __restrict__ xz, const float* __restrict__ cw,
                              const float* __restrict__ cb, float* __restrict__ xs) {
    int i = blockIdx.x * blockDim.x + threadIdx.x;
    if (i >= L_TOK * DI) return;
    int l = i / DI, d = i % DI;
    float s = cb[d];
#pragma unroll
    for (int k = 0; k < 4; ++k) {
        int ls = l + k - 3;
        if (ls >= 0) s += xz[(size_t)ls * 128 + d] * cw[d * 4 + k];
    }
    xs[i] = siluf(s);
}

// dt = softplus(xdbl[:, :2] @ dt_w^T + dt_b)
__global__ void dt_k(const float* __restrict__ xdbl, const float* __restrict__ dw,
                     const float* __restrict__ db, float* __restrict__ dt) {
    int i = blockIdx.x * blockDim.x + threadIdx.x;
    if (i >= L_TOK * DI) return;
    int l = i / DI, d = i % DI;
    float v = xdbl[(size_t)l * LDX] * dw[d * 2] + xdbl[(size_t)l * LDX + 1] * dw[d * 2 + 1] + db[d];
    dt[i] = softplusf(v);
}

// chunked selective scan, pass 1: per-(chunk, channel) transition (prodA, partial h)
__global__ void scan1_k(const float* __restrict__ dt, const float* __restrict__ xs,
                        const float* __restrict__ xdbl, const float* __restrict__ Alog,
                        float* __restrict__ chA, float* __restrict__ chB) {
    int c = blockIdx.x, d = threadIdx.x;
    float aS[DS], aP[DS], hS[DS];
#pragma unroll
    for (int s = 0; s < DS; ++s) { aS[s] = -__expf(Alog[d * DS + s]); aP[s] = 1.0f; hS[s] = 0.0f; }
    int l0 = c * CH;
    for (int li = 0; li < CH; ++li) {
        int l = l0 + li;
        float dtv = dt[(size_t)l * DI + d];
        float du = dtv * xs[(size_t)l * DI + d];
        const float* Br = xdbl + (size_t)l * LDX + DTR;
#pragma unroll
        for (int s = 0; s < DS; ++s) {
            float dA = __expf(dtv * aS[s]);
            hS[s] = dA * hS[s] + du * Br[s];
            aP[s] *= dA;
        }
    }
    size_t base = ((size_t)c * DI + d) * DS;
#pragma unroll
    for (int s = 0; s < DS; ++s) { chA[base + s] = aP[s]; chB[base + s] = hS[s]; }
}

// pass 2: sequential chunk prefix (tiny: 448 steps, 64 threads)
__global__ void scan2_k(const float* __restrict__ chA, const float* __restrict__ chB,
                        float* __restrict__ hIn) {
    int d = threadIdx.x;
    float h[DS];
#pragma unroll
    for (int s = 0; s < DS; ++s) h[s] = 0.0f;
    for (int c = 0; c < NC; ++c) {
        size_t base = ((size_t)c * DI + d) * DS;
#pragma unroll
        for (int s = 0; s < DS; ++s) hIn[base + s] = h[s];
#pragma unroll
        for (int s = 0; s < DS; ++s) h[s] = chA[base + s] * h[s] + chB[base + s];
    }
}

// pass 3: replay with true initial state, fused gating: y = (scan + xs*D) * silu(z)
__global__ void scan3_k(const float* __restrict__ dt, const float* __restrict__ xs,
                        const float* __restrict__ xdbl, const float* __restrict__ Alog,
                        const float* __restrict__ hIn, const float* __restrict__ Dp,
                        const float* __restrict__ xz, float* __restrict__ y) {
    int c = blockIdx.x, d = threadIdx.x;
    float aS[DS], h[DS];
    size_t base = ((size_t)c * DI + d) * DS;
#pragma unroll
    for (int s = 0; s < DS; ++s) { aS[s] = -__expf(Alog[d * DS + s]); h[s] = hIn[base + s]; }
    float Dv = Dp[d];
    int l0 = c * CH;
    for (int li = 0; li < CH; ++li) {
        int l = l0 + li;
        float dtv = dt[(size_t)l * DI + d];
        float xv = xs[(size_t)l * DI + d];
        float du = dtv * xv;
        const float* Br = xdbl + (size_t)l * LDX + DTR;
        const float* Cr = xdbl + (size_t)l * LDX + DTR + DS;
        float yv = 0.0f;
#pragma unroll
        for (int s = 0; s < DS; ++s) {
            float dA = __expf(dtv * aS[s]);
            h[s] = dA * h[s] + du * Br[s];
            yv += h[s] * Cr[s];
        }
        float z = xz[(size_t)l * 128 + DI + d];
        y[(size_t)l * DI + d] = (yv + xv * Dv) * siluf(z);
    }
}

// ---------------- host-side orchestration ----------------
struct MambaP { const float *A_log, *D, *conv_b, *conv_w, *dt_b, *dt_w, *in_w, *out_w, *xproj_w; };
struct HinP { const float *b1, *b2, *nb, *nw, *w1, *w2; };
struct EncP { HinP blk[3]; const float *cb, *cw; };

static MambaP mk_mamba(void* const* din, int b) {
    MambaP m;
    m.A_log   = (const float*)din[b + 0];
    m.D       = (const float*)din[b + 1];
    m.conv_b  = (const float*)din[b + 2];
    m.conv_w  = (const float*)din[b + 3];
    m.dt_b    = (const float*)din[b + 4];
    m.dt_w    = (const float*)din[b + 5];
    m.in_w    = (const float*)din[b + 6];
    m.out_w   = (const float*)din[b + 7];
    m.xproj_w = (const float*)din[b + 8];
    return m;
}
static EncP mk_enc(void* const* din, int b) {  // b = index of blocks[0].b1
    EncP e;
    for (int i = 0; i < 3; ++i) {
        int o = b + i * 6;
        e.blk[i].b1 = (const float*)din[o + 0];
        e.blk[i].b2 = (const float*)din[o + 1];
        e.blk[i].nb = (const float*)din[o + 2];
        e.blk[i].nw = (const float*)din[o + 3];
        e.blk[i].w1 = (const float*)din[o + 4];
        e.blk[i].w2 = (const float*)din[o + 5];
    }
    e.cb = (const float*)din[b + 18];
    e.cw = (const float*)din[b + 19];
    return e;
}

struct Scratch {
    float *featA, *featB, *featC;
    float *msTok, *panTok, *resMs, *resPan, *normA, *normB, *mmInA, *mmInB;
    float *xz, *xs, *xdbl, *dt, *y;
    float *chA, *chB, *hIn, *headAcc;
};

static void gemm(const float* A, int lda, int aT, const float* W, const float* bias,
                 float* C, int ldc, int M, int N, int K, hipStream_t s) {
    int tiles = (M >> 4) * ((N + 15) >> 4);
    int blocks = (tiles + 7) / 8;
    if (aT) gemm_wmma_k<1><<<blocks, 256, 0, s>>>(A, lda, W, bias, C, ldc, M, N, K);
    else    gemm_wmma_k<0><<<blocks, 256, 0, s>>>(A, lda, W, bias, C, ldc, M, N, K);
}

static void run_mamba(const float* in, const MambaP& m, float* out, Scratch& w, hipStream_t s) {
    int nLD = L_TOK * DI, eb = (nLD + 255) / 256;
    gemm(in, DM, 0, m.in_w, nullptr, w.xz, 128, L_TOK, 2 * DI, DM, s);          // in_proj
    conv1d_silu_k<<<eb, 256, 0, s>>>(w.xz, m.conv_w, m.conv_b, w.xs);           // dwconv + silu
    gemm(w.xs, DI, 0, m.xproj_w, nullptr, w.xdbl, LDX, L_TOK, NX, DI, s);       // x_proj
    dt_k<<<eb, 256, 0, s>>>(w.xdbl, m.dt_w, m.dt_b, w.dt);                      // dt_proj+softplus
    scan1_k<<<NC, DI, 0, s>>>(w.dt, w.xs, w.xdbl, m.A_log, w.chA, w.chB);
    scan2_k<<<1, DI, 0, s>>>(w.chA, w.chB, w.hIn);
    scan3_k<<<NC, DI, 0, s>>>(w.dt, w.xs, w.xdbl, m.A_log, w.hIn, m.D, w.xz, w.y);
    gemm(w.y, DI, 0, m.out_w, nullptr, out, DM, L_TOK, DM, DI, s);              // out_proj
}

static void run_encoder(const float* img, int cin, const EncP& e, Scratch& w, hipStream_t s) {
    int n = DM * L_TOK, b = (n + 255) / 256;
    conv3_k<<<b, 256, 0, s>>>(img, cin, e.cw, e.cb, w.featA, 0);
    for (int i = 0; i < 3; ++i) {
        conv3_k<<<b, 256, 0, s>>>(w.featA, DM, e.blk[i].w1, e.blk[i].b1, w.featB, 1);
        instnorm_k<<<16, 256, 0, s>>>(w.featB, e.blk[i].nw, e.blk[i].nb);
        conv3_k<<<b, 256, 0, s>>>(w.featB, DM, e.blk[i].w2, e.blk[i].b2, w.featC, 1);
        addip_k<<<b, 256, 0, s>>>(w.featA, w.featC, n);
    }
}

extern "C" void kernel_launch(void* const* d_in, const int* in_sizes, int n_in,
                              void* d_out, int out_size, void* d_ws, size_t ws_size,
                              hipStream_t stream) {
    (void)in_sizes; (void)n_in; (void)out_size; (void)ws_size;
    const float* ms = (const float*)d_in[0];
    const float* pan = (const float*)d_in[1];

    // jax pytree flattening (dict keys sorted, lists in order):
    // 2..23 ms_blocks, 24..43 ms_enc, 44..45 ms_tok, 46 out_b, 47 out_w,
    // 48..69 pan_blocks, 70..89 pan_enc, 90..91 pan_tok, 92..113 swap1, 114..135 swap2
    const float* msTokB = (const float*)d_in[44];
    const float* msTokW = (const float*)d_in[45];
    const float* outB   = (const float*)d_in[46];
    const float* outW   = (const float*)d_in[47];
    const float* panTokB = (const float*)d_in[90];
    const float* panTokW = (const float*)d_in[91];

    // ---- carve workspace ----
    float* p = (float*)d_ws;
    Scratch w;
    auto take = [&](size_t n) { float* r = p; p += n; return r; };
    w.featA = take((size_t)DM * L_TOK); w.featB = take((size_t)DM * L_TOK); w.featC = take((size_t)DM * L_TOK);
    w.msTok = take((size_t)L_TOK * DM); w.panTok = take((size_t)L_TOK * DM);
    w.resMs = take((size_t)L_TOK * DM); w.resPan = take((size_t)L_TOK * DM);
    w.normA = take((size_t)L_TOK * DM); w.normB = take((size_t)L_TOK * DM);
    w.mmInA = take((size_t)L_TOK * DM); w.mmInB = take((size_t)L_TOK * DM);
    w.xz = take((size_t)L_TOK * 128); w.xs = take((size_t)L_TOK * DI);
    w.xdbl = take((size_t)L_TOK * LDX); w.dt = take((size_t)L_TOK * DI); w.y = take((size_t)L_TOK * DI);
    w.chA = take((size_t)NC * DI * DS); w.chB = take((size_t)NC * DI * DS); w.hIn = take((size_t)NC * DI * DS);
    w.headAcc = take((size_t)2 * DM * OUTD);

    int nTok = L_TOK * DM, tb = (nTok + 255) / 256;
    int lnB = (L_TOK + 255) / 256;

    // ---- encoders + patch embed (WMMA GEMM, A channel-major -> AT=1) ----
    run_encoder(ms, 4, mk_enc(d_in, 24), w, stream);
    gemm(w.featA, L_TOK, 1, msTokW, msTokB, w.msTok, DM, L_TOK, DM, DM, stream);
    run_encoder(pan, 4, mk_enc(d_in, 70), w, stream);
    gemm(w.featA, L_TOK, 1, panTokW, panTokB, w.panTok, DM, L_TOK, DM, DM, stream);

    // ---- SingleMambaBlock chains ----
    fill_k<<<tb, 256, 0, stream>>>(w.resMs, nTok, 0.0f);
    fill_k<<<tb, 256, 0, stream>>>(w.resPan, nTok, 0.0f);
    for (int j = 0; j < 2; ++j) {
        int b = 2 + 11 * j;  // ms_blocks[j]: ln_b, ln_w, mamba(9)
        addip_k<<<tb, 256, 0, stream>>>(w.resMs, w.msTok, nTok);
        layernorm_k<<<lnB, 256, 0, stream>>>(w.resMs, (const float*)d_in[b + 1], (const float*)d_in[b + 0], w.normA);
        MambaP m = mk_mamba(d_in, b + 2);
        run_mamba(w.normA, m, w.msTok, w, stream);
    }
    for (int j = 0; j < 2; ++j) {
        int b = 48 + 11 * j;  // pan_blocks[j]
        addip_k<<<tb, 256, 0, stream>>>(w.resPan, w.panTok, nTok);
        layernorm_k<<<lnB, 256, 0, stream>>>(w.resPan, (const float*)d_in[b + 1], (const float*)d_in[b + 0], w.normA);
        MambaP m = mk_mamba(d_in, b + 2);
        run_mamba(w.normA, m, w.panTok, w, stream);
    }

    // ---- TokenSwapMamba x2 ----
    for (int sw = 0; sw < 2; ++sw) {
        int b = 92 + 22 * sw;  // ln1_b, ln1_w, ln2_b, ln2_w, ms_mamba(9), pan_mamba(9)
        addip_k<<<tb, 256, 0, stream>>>(w.resMs, w.msTok, nTok);
        addip_k<<<tb, 256, 0, stream>>>(w.resPan, w.panTok, nTok);
        layernorm_k<<<lnB, 256, 0, stream>>>(w.resMs, (const float*)d_in[b + 1], (const float*)d_in[b + 0], w.normA);   // msn
        layernorm_k<<<lnB, 256, 0, stream>>>(w.resPan, (const float*)d_in[b + 3], (const float*)d_in[b + 2], w.normB);  // pann
        concat_k<<<tb, 256, 0, stream>>>(w.mmInA, w.normB, w.normA);  // ms in:  lo=pann, hi=msn
        concat_k<<<tb, 256, 0, stream>>>(w.mmInB, w.normA, w.normB);  // pan in: lo=msn,  hi=pann
        MambaP mm = mk_mamba(d_in, b + 4);
        MambaP pm = mk_mamba(d_in, b + 13);
        run_mamba(w.mmInA, mm, w.msTok, w, stream);
        run_mamba(w.mmInB, pm, w.panTok, w, stream);
    }

    // ---- shared head: relu(tok^T @ out_w^T + b), then per-modality channel means ----
    float* out = (float*)d_out;                       // [ms_o | pan_o | ms_feat | pan_feat]
    int accN = 2 * DM * OUTD;
    fill_k<<<(accN + 255) / 256, 256, 0, stream>>>(w.headAcc, accN, 0.0f);
    head_gemm_k<<<NHB * KSPL, 256, 0, stream>>>(w.msTok, w.panTok, outW, w.headAcc);
    head_epi_k<<<(accN + 255) / 256, 256, 0, stream>>>(w.headAcc, outB, out);
    head_feat_k<<<(2 * OUTD + 255) / 256, 256, 0, stream>>>(out, out + (size_t)accN);
}